// WalkPositionEncoder_6665789243524
// MI455X (gfx1250) — compile-verified
//
#include <hip/hip_runtime.h>

// WalkPositionEncoder for MI455X (gfx1250).
// B=256 batches, K=64 walks, L=32 steps, SENTINEL=32, HALF=64 (row = 128 f32).
// Store-bound kernel (512MB out). Strategy: one workgroup per batch; LDS hash
// table replaces the reference's 205MB scatter-min table; walks + embedding
// tables staged into LDS via CDNA5 async global->LDS loads; output written as
// one 512B coalesced burst per row by a full wave.

#define BATCHES   256
#define KWALKS    64
#define LSTEPS    32
#define M_ELEMS   (KWALKS * LSTEPS)   // 2048
#define SENT      32u
#define HSIZE     4096
#define HMASK     (HSIZE - 1)
#define EMBROWS   33                  // L+1
#define HALFD     64                  // floats per half-row
#define NTHREADS  512

// ---- CDNA5 async global->LDS copy (inline asm; bypasses toolchain builtin
// arity differences). vdst = 32-bit LDS byte address, vaddr = 64-bit global. ----
__device__ __forceinline__ unsigned lds_off(const void* p) {
    // Flat LDS addresses carry the LDS byte offset in addr[31:0].
    return (unsigned)(unsigned long long)p;
}

__device__ __forceinline__ void async_load_b128(unsigned lds_addr, const void* gaddr) {
    asm volatile("global_load_async_to_lds_b128 %0, %1, off"
                 :: "v"(lds_addr), "v"(gaddr) : "memory");
}

__device__ __forceinline__ void wait_async0() {
#if defined(__has_builtin)
#if __has_builtin(__builtin_amdgcn_s_wait_asynccnt)
    __builtin_amdgcn_s_wait_asynccnt(0);
    return;
#endif
#endif
    asm volatile("s_wait_asynccnt 0" ::: "memory");
}

// ---- open-addressing hash: key==0 means empty (node id 0 is never valid) ----
__device__ __forceinline__ unsigned hslot(unsigned node) {
    return (node * 2654435761u) >> 20;   // top 12 bits -> [0,4095]
}

__device__ __forceinline__ void insert_tbl(unsigned* keys, unsigned* vals,
                                           unsigned node, unsigned pos) {
    unsigned s = hslot(node);
    for (;;) {
        unsigned prev = atomicCAS(&keys[s], 0u, node);
        if (prev == 0u || prev == node) { atomicMin(&vals[s], pos); return; }
        s = (s + 1) & HMASK;
    }
}

__device__ __forceinline__ unsigned probe_tbl(const unsigned* keys, const unsigned* vals,
                                              unsigned node) {
    unsigned s = hslot(node);
    for (;;) {
        unsigned kk = keys[s];
        if (kk == node) return vals[s];
        if (kk == 0u)   return SENT;
        s = (s + 1) & HMASK;
    }
}

__global__ __launch_bounds__(NTHREADS)
void wpe_kernel(const int* __restrict__ src_walks, const int* __restrict__ tgt_walks,
                const int* __restrict__ src_lens, const int* __restrict__ tgt_lens,
                const float* __restrict__ own_emb, const float* __restrict__ cross_emb,
                float* __restrict__ out) {
    __shared__ __align__(16) unsigned nodes[2][M_ELEMS];        // 16 KB
    __shared__ unsigned keys[2][HSIZE];                          // 32 KB
    __shared__ unsigned vals[2][HSIZE];                          // 32 KB
    __shared__ __align__(16) float4 embs[2][EMBROWS * 16];       // ~16.9 KB (own, cross)
    __shared__ unsigned packedIdx[2][M_ELEMS];                   // 16 KB
    __shared__ int lens_s[2][KWALKS];                            // 0.5 KB

    const int b   = blockIdx.x;
    const int tid = threadIdx.x;

    // ---- Phase 0: issue async copies (walks + embedding tables -> LDS) ----
    {
        const int* gs = src_walks + (size_t)b * M_ELEMS;
        const int* gt = tgt_walks + (size_t)b * M_ELEMS;
        // 2048 ints = 512 b128 chunks per table; exactly one per thread.
        async_load_b128(lds_off(&nodes[0][tid * 4]), gs + tid * 4);
        async_load_b128(lds_off(&nodes[1][tid * 4]), gt + tid * 4);
        // 33*64 f32 = 528 float4 chunks per table.
        const float4* ge0 = (const float4*)own_emb;
        const float4* ge1 = (const float4*)cross_emb;
        for (int i = tid; i < EMBROWS * 16; i += NTHREADS) {
            async_load_b128(lds_off(&embs[0][i]), ge0 + i);
            async_load_b128(lds_off(&embs[1][i]), ge1 + i);
        }
    }

    // ---- overlap: init hash tables + copy lens while async copies fly ----
    for (int i = tid; i < HSIZE; i += NTHREADS) {
        keys[0][i] = 0u; keys[1][i] = 0u;
        vals[0][i] = SENT; vals[1][i] = SENT;
    }
    if (tid < KWALKS) {
        lens_s[0][tid] = src_lens[b * KWALKS + tid];
        lens_s[1][tid] = tgt_lens[b * KWALKS + tid];
    }

    wait_async0();
    __syncthreads();

    // ---- Phase 1: build scatter-min tables (src=0, tgt=1) ----
    #pragma unroll
    for (int t = 0; t < 2; ++t) {
        for (int i = tid; i < M_ELEMS; i += NTHREADS) {
            unsigned node = nodes[t][i];
            int l = i & (LSTEPS - 1);
            int k = i >> 5;
            if (node != 0u && l < lens_s[t][k])
                insert_tbl(keys[t], vals[t], node, (unsigned)l);
        }
    }
    __syncthreads();

    // ---- Phase 2: per-element own/cross first-seen positions ----
    #pragma unroll
    for (int t = 0; t < 2; ++t) {
        for (int i = tid; i < M_ELEMS; i += NTHREADS) {
            unsigned node = nodes[t][i];
            int l = i & (LSTEPS - 1);
            int k = i >> 5;
            bool valid = (node != 0u) && (l < lens_s[t][k]);
            unsigned own = SENT, cross = SENT;
            if (valid) {
                own   = probe_tbl(keys[t],     vals[t],     node);
                cross = probe_tbl(keys[1 - t], vals[1 - t], node);
            }
            packedIdx[t][i] = own | (cross << 8) | (valid ? 0x10000u : 0u);
        }
    }
    __syncthreads();

    // ---- Phase 3: write output. One wave per 128-float row (512B burst). ----
    // Row layout: src tensor rows [0, B*M), then tgt tensor rows.
    const int lane = tid & 31;
    const int wave = tid >> 5;              // 16 waves
    float4* outv = (float4*)out;
    const float4 zero4 = make_float4(0.f, 0.f, 0.f, 0.f);

    for (int e = wave; e < 2 * M_ELEMS; e += (NTHREADS / 32)) {
        const int t = e >> 11;              // 0 = src, 1 = tgt
        const int i = e & (M_ELEMS - 1);
        const unsigned p = packedIdx[t][i]; // LDS broadcast (uniform per wave)
        const long long row = (long long)t * ((long long)BATCHES * M_ELEMS)
                            + (long long)b * M_ELEMS + i;
        float4* dst = outv + row * 32 + lane;   // 128 floats = 32 float4 per row
        if (p & 0x10000u) {
            const int half = lane >> 4;         // lanes 0-15 own, 16-31 cross
            const unsigned pos = (half == 0) ? (p & 0xFFu) : ((p >> 8) & 0xFFu);
            *dst = embs[half][pos * 16 + (lane & 15)];
        } else {
            *dst = zero4;
        }
    }
}

extern "C" void kernel_launch(void* const* d_in, const int* in_sizes, int n_in,
                              void* d_out, int out_size, void* d_ws, size_t ws_size,
                              hipStream_t stream) {
    const int*   src_walks = (const int*)d_in[0];
    const int*   tgt_walks = (const int*)d_in[1];
    const int*   src_lens  = (const int*)d_in[2];
    const int*   tgt_lens  = (const int*)d_in[3];
    const float* own_emb   = (const float*)d_in[4];
    const float* cross_emb = (const float*)d_in[5];
    float*       out       = (float*)d_out;
    (void)in_sizes; (void)n_in; (void)out_size; (void)d_ws; (void)ws_size;

    dim3 grid(BATCHES), block(NTHREADS);
    wpe_kernel<<<grid, block, 0, stream>>>(src_walks, tgt_walks, src_lens, tgt_lens,
                                           own_emb, cross_emb, out);
}